// PlayerPerformanceGNN_60026462929065
// MI455X (gfx1250) — compile-verified
//
#include <hip/hip_runtime.h>

// ---------------------------------------------------------------------------
// GCN forward for MI455X (gfx1250, wave32).
// Memory-bound scatter/gather problem; dense GEMMs done with exact-f32
// V_WMMA_F32_16X16X4_F32 (one 16-row tile per wave, chained accumulators).
// Edge normalization computed on the fly from L2-resident dinv table.
// ---------------------------------------------------------------------------

typedef float v2f __attribute__((ext_vector_type(2)));
typedef float v8f __attribute__((ext_vector_type(8)));

#define N_NODES   1000000
#define N_EDGES   16000000
#define F_IN      32
#define F_H1      16
#define F_H2      8
#define N_CLS     5
#define N_GRAPHS  10000

#define ATOMIC_FADD(p, v) \
  __hip_atomic_fetch_add((p), (v), __ATOMIC_RELAXED, __HIP_MEMORY_SCOPE_AGENT)

// ------------------------------ utility ------------------------------------

__global__ void zero_f32(float* __restrict__ p, long long n) {
  long long i = (long long)blockIdx.x * blockDim.x + threadIdx.x;
  long long stride = (long long)gridDim.x * blockDim.x;
  for (; i < n; i += stride) p[i] = 0.0f;
}

// deg[dst] += 1 for every edge (self-loop +1 added in rsqrt pass)
__global__ void degree_kernel(const int* __restrict__ dst,
                              float* __restrict__ deg, int nE) {
  int e = blockIdx.x * blockDim.x + threadIdx.x;
  if (e < nE) ATOMIC_FADD(&deg[dst[e]], 1.0f);
}

// deg -> deg_inv_sqrt in place (A_hat = A + I, so +1)
__global__ void rsqrt_kernel(float* __restrict__ deg, int n) {
  int i = blockIdx.x * blockDim.x + threadIdx.x;
  if (i < n) deg[i] = rsqrtf(deg[i] + 1.0f);
}

// ------------------------------ WMMA GEMMs ---------------------------------
// One wave32 computes a 16-row tile: C[16x16] = A[16xK] * B[Kx16] via chained
// V_WMMA_F32_16X16X4_F32.  A layout (32-bit, 16x4): lanes 0-15 hold M=lane,
// VGPR0/1 = K,K+1; lanes 16-31 hold M=lane-16, VGPR0/1 = K+2,K+3.
// C/D layout: lane holds column N=lane&15; VGPR j holds row M = j + 8*(lane>>4).

__global__ __launch_bounds__(128) void gemm_x_w1(const float* __restrict__ X,
                                                 const float* __restrict__ W1,
                                                 float* __restrict__ H1pre) {
  int wave = blockIdx.x * (blockDim.x >> 5) + (threadIdx.x >> 5);
  int lane = threadIdx.x & 31;
  int m  = lane & 15;            // row within tile (A) / column (B,C)
  int kk = (lane >> 4) << 1;     // 0 or 2: K-pair this half-wave owns
  const float* xrow = X + (size_t)(wave * 16 + m) * F_IN;

  v8f c = {0.f, 0.f, 0.f, 0.f, 0.f, 0.f, 0.f, 0.f};
#pragma unroll
  for (int k = 0; k < F_IN; k += 4) {
    v2f a, b;
    a.x = xrow[k + kk];
    a.y = xrow[k + kk + 1];
    b.x = W1[(k + kk) * F_H1 + m];       // B[K][N], N = lane&15
    b.y = W1[(k + kk + 1) * F_H1 + m];
    c = __builtin_amdgcn_wmma_f32_16x16x4_f32(false, a, false, b,
                                              (short)0, c, false, false);
  }
  int hi = lane >> 4;
  float* out = H1pre + (size_t)(wave * 16 + 8 * hi) * F_H1 + m;
#pragma unroll
  for (int j = 0; j < 8; ++j) out[(size_t)j * F_H1] = c[j];
}

__global__ __launch_bounds__(128) void gemm_h1_w2(const float* __restrict__ H1,
                                                  const float* __restrict__ W2,
                                                  float* __restrict__ H2pre) {
  int wave = blockIdx.x * (blockDim.x >> 5) + (threadIdx.x >> 5);
  int lane = threadIdx.x & 31;
  int m  = lane & 15;
  int kk = (lane >> 4) << 1;
  const float* hrow = H1 + (size_t)(wave * 16 + m) * F_H1;
  bool live_col = (m < F_H2);    // W2 is 16x8: pad columns 8..15 with zeros

  v8f c = {0.f, 0.f, 0.f, 0.f, 0.f, 0.f, 0.f, 0.f};
#pragma unroll
  for (int k = 0; k < F_H1; k += 4) {
    v2f a, b;
    a.x = hrow[k + kk];
    a.y = hrow[k + kk + 1];
    b.x = live_col ? W2[(k + kk) * F_H2 + m] : 0.0f;
    b.y = live_col ? W2[(k + kk + 1) * F_H2 + m] : 0.0f;
    c = __builtin_amdgcn_wmma_f32_16x16x4_f32(false, a, false, b,
                                              (short)0, c, false, false);
  }
  if (live_col) {
    int hi = lane >> 4;
    float* out = H2pre + (size_t)(wave * 16 + 8 * hi) * F_H2 + m;
#pragma unroll
    for (int j = 0; j < 8; ++j) out[(size_t)j * F_H2] = c[j];
  }
}

// --------------------------- edge aggregation ------------------------------
// msg = h[src] * (dinv[src]*dinv[dst]); agg[dst] += msg.  dinv is a 4MB
// L2-resident table, so norm_e is never materialized (saves 192MB of traffic).

__global__ void agg_edges16(const int* __restrict__ src, const int* __restrict__ dst,
                            const float* __restrict__ dinv,
                            const float* __restrict__ H, float* __restrict__ AGG,
                            int nE) {
  int e = blockIdx.x * blockDim.x + threadIdx.x;
  if (e >= nE) return;
  int s = src[e], d = dst[e];
  float w = dinv[s] * dinv[d];
  const float4* hs = (const float4*)(H + (size_t)s * F_H1);
  float* out = AGG + (size_t)d * F_H1;
#pragma unroll
  for (int q = 0; q < 4; ++q) {
    float4 v = hs[q];
    ATOMIC_FADD(&out[4 * q + 0], v.x * w);
    ATOMIC_FADD(&out[4 * q + 1], v.y * w);
    ATOMIC_FADD(&out[4 * q + 2], v.z * w);
    ATOMIC_FADD(&out[4 * q + 3], v.w * w);
  }
}

__global__ void agg_edges8(const int* __restrict__ src, const int* __restrict__ dst,
                           const float* __restrict__ dinv,
                           const float* __restrict__ H, float* __restrict__ AGG,
                           int nE) {
  int e = blockIdx.x * blockDim.x + threadIdx.x;
  if (e >= nE) return;
  int s = src[e], d = dst[e];
  float w = dinv[s] * dinv[d];
  const float4* hs = (const float4*)(H + (size_t)s * F_H2);
  float* out = AGG + (size_t)d * F_H2;
#pragma unroll
  for (int q = 0; q < 2; ++q) {
    float4 v = hs[q];
    ATOMIC_FADD(&out[4 * q + 0], v.x * w);
    ATOMIC_FADD(&out[4 * q + 1], v.y * w);
    ATOMIC_FADD(&out[4 * q + 2], v.z * w);
    ATOMIC_FADD(&out[4 * q + 3], v.w * w);
  }
}

// --------------------------- layer finalize --------------------------------
// agg = relu(agg + hpre * dinv^2 + b)  (in place; agg becomes the layer output)

__global__ void finalize1(float* __restrict__ agg, const float* __restrict__ hpre,
                          const float* __restrict__ dinv,
                          const float* __restrict__ b1, int n) {
  int i = blockIdx.x * blockDim.x + threadIdx.x;
  if (i >= n) return;
  float w = dinv[i]; w *= w;
  float4* a = (float4*)(agg + (size_t)i * F_H1);
  const float4* h = (const float4*)(hpre + (size_t)i * F_H1);
  const float4* b = (const float4*)b1;
#pragma unroll
  for (int q = 0; q < 4; ++q) {
    float4 av = a[q], hv = h[q], bv = b[q];
    av.x = fmaxf(av.x + hv.x * w + bv.x, 0.0f);
    av.y = fmaxf(av.y + hv.y * w + bv.y, 0.0f);
    av.z = fmaxf(av.z + hv.z * w + bv.z, 0.0f);
    av.w = fmaxf(av.w + hv.w * w + bv.w, 0.0f);
    a[q] = av;
  }
}

// finalize layer 2 + fused global-mean-pool accumulation
__global__ void finalize2_pool(const float* __restrict__ agg,
                               const float* __restrict__ hpre,
                               const float* __restrict__ dinv,
                               const float* __restrict__ b2,
                               const int* __restrict__ batch,
                               float* __restrict__ pooled,
                               float* __restrict__ cnt, int n) {
  int i = blockIdx.x * blockDim.x + threadIdx.x;
  if (i >= n) return;
  float w = dinv[i]; w *= w;
  int g = batch[i];
  float* ps = pooled + (size_t)g * F_H2;
#pragma unroll
  for (int f = 0; f < F_H2; ++f) {
    float v = agg[(size_t)i * F_H2 + f] + hpre[(size_t)i * F_H2 + f] * w + b2[f];
    v = fmaxf(v, 0.0f);
    ATOMIC_FADD(&ps[f], v);
  }
  ATOMIC_FADD(&cnt[g], 1.0f);
}

// readout: out[g] = (pooled[g]/max(cnt,1)) @ Wl + bl
__global__ void readout(const float* __restrict__ pooled, const float* __restrict__ cnt,
                        const float* __restrict__ Wl, const float* __restrict__ bl,
                        float* __restrict__ out, int G) {
  int g = blockIdx.x * blockDim.x + threadIdx.x;
  if (g >= G) return;
  float inv = 1.0f / fmaxf(cnt[g], 1.0f);
  float p[F_H2];
#pragma unroll
  for (int f = 0; f < F_H2; ++f) p[f] = pooled[(size_t)g * F_H2 + f] * inv;
#pragma unroll
  for (int c = 0; c < N_CLS; ++c) {
    float acc = bl[c];
#pragma unroll
    for (int f = 0; f < F_H2; ++f) acc += p[f] * Wl[f * N_CLS + c];
    out[(size_t)g * N_CLS + c] = acc;
  }
}

// ------------------------------ launcher -----------------------------------

extern "C" void kernel_launch(void* const* d_in, const int* in_sizes, int n_in,
                              void* d_out, int out_size, void* d_ws, size_t ws_size,
                              hipStream_t stream) {
  const float* x     = (const float*)d_in[0];
  const int*   ei    = (const int*)d_in[1];
  const int*   src   = ei;                // edge_index[0]
  const int*   dst   = ei + N_EDGES;      // edge_index[1]
  const int*   batch = (const int*)d_in[2];
  // d_in[3] = num_graphs scalar (value known at compile time: N_GRAPHS)
  const float* W1 = (const float*)d_in[4];
  const float* b1 = (const float*)d_in[5];
  const float* W2 = (const float*)d_in[6];
  const float* b2 = (const float*)d_in[7];
  const float* Wl = (const float*)d_in[8];
  const float* bl = (const float*)d_in[9];
  float* out = (float*)d_out;

  // workspace layout (bytes):   total ~ 132.3 MB
  //   [0,   4MB)  dinv (1M f32)
  //   [4MB, 68MB) bufA: h1_pre (64MB) -> later h2_pre (first 32MB) + agg2 (32MB @ +32MB)
  //   [68MB,132MB) bufB: agg1 -> h1 (in place)
  //   [132MB, ..) pooled (10000*8 f32) then cnt (10000 f32)
  char* ws = (char*)d_ws;
  float* dinv   = (float*)(ws);
  float* bufA   = (float*)(ws + (4ll  << 20));
  float* bufB   = (float*)(ws + (68ll << 20));
  float* pooled = (float*)(ws + (132ll << 20));
  float* cnt    = pooled + (size_t)N_GRAPHS * F_H2;

  float* h1pre = bufA;
  float* agg1  = bufB;                       // becomes h1 after finalize1
  float* h2pre = bufA;                       // reuse (h1pre dead after finalize1)
  float* agg2  = (float*)((char*)bufA + (32ll << 20));

  const int EB = 256;
  const int EG = (N_EDGES + EB - 1) / EB;
  const int NB = 256;
  const int NG = (N_NODES + NB - 1) / NB;
  const int GEMM_BLOCKS = N_NODES / 64;      // 4 waves/block, 16 rows/wave

  // degrees -> dinv
  zero_f32<<<2048, 256, 0, stream>>>(dinv, (long long)N_NODES);
  degree_kernel<<<EG, EB, 0, stream>>>(dst, dinv, N_EDGES);
  rsqrt_kernel<<<NG, NB, 0, stream>>>(dinv, N_NODES);

  // layer 1
  gemm_x_w1<<<GEMM_BLOCKS, 128, 0, stream>>>(x, W1, h1pre);
  zero_f32<<<8192, 256, 0, stream>>>(agg1, (long long)N_NODES * F_H1);
  agg_edges16<<<EG, EB, 0, stream>>>(src, dst, dinv, h1pre, agg1, N_EDGES);
  finalize1<<<NG, NB, 0, stream>>>(agg1, h1pre, dinv, b1, N_NODES);

  // layer 2
  gemm_h1_w2<<<GEMM_BLOCKS, 128, 0, stream>>>(agg1, W2, h2pre);
  zero_f32<<<8192, 256, 0, stream>>>(agg2, (long long)N_NODES * F_H2);
  agg_edges8<<<EG, EB, 0, stream>>>(src, dst, dinv, h2pre, agg2, N_EDGES);

  // pooling + readout
  zero_f32<<<128, 256, 0, stream>>>(pooled, (long long)N_GRAPHS * (F_H2 + 1));
  finalize2_pool<<<NG, NB, 0, stream>>>(agg2, h2pre, dinv, b2, batch,
                                        pooled, cnt, N_NODES);
  readout<<<(N_GRAPHS + NB - 1) / NB, NB, 0, stream>>>(pooled, cnt, Wl, bl,
                                                       out, N_GRAPHS);
}